// LinearSiso_9096740733409
// MI455X (gfx1250) — compile-verified
//
#include <hip/hip_runtime.h>

typedef __attribute__((ext_vector_type(2))) float v2f;
typedef __attribute__((ext_vector_type(8))) float v8f;

// Branchless: all lanes compute h[0..16] fully unrolled (impulse response of
// y[t] = -a1*y[t-1] - a2*y[t-2]); lane n captures h[n] via a cndmask chain.
__device__ __forceinline__ float h_capture_lane(float a1, float a2, int lane) {
    float h[17];
    h[0] = 1.0f;
    h[1] = -a1;
#pragma unroll
    for (int i = 2; i <= 16; ++i) h[i] = -a1 * h[i - 1] - a2 * h[i - 2];
    float hl = h[0];
#pragma unroll
    for (int i = 1; i <= 16; ++i) hl = (lane == i) ? h[i] : hl;
    return hl;   // lane n (n<=16) now holds h[n]
}

__device__ __forceinline__ float h_gather(float hl, int n) {
    // gather h[n] from lane n via DS lane-permute; n < 0 -> 0 (upper triangle)
    int nn = n < 0 ? 0 : n;
    float v = __int_as_float(__builtin_amdgcn_ds_bpermute(nn << 2, __float_as_int(hl)));
    return n < 0 ? 0.0f : v;
}

// Kernel 1: per 16(batch)x16(time) tile, zero-state response via fp32 WMMA.
//   Y0[m, j] = sum_k X[m, k] * h[j - k]   (16x16x16 GEMM = 4 chained 16x16x4 f32 WMMAs)
// One wave per tile; u-tile staged through LDS (coalesced loads -> A layout).
__global__ void __launch_bounds__(256)
iir_tile_fir_wmma(const float* __restrict__ u,
                  const float* __restrict__ u_init,
                  const float* __restrict__ bc,
                  const float* __restrict__ ac,
                  float* __restrict__ y0,
                  int T, int tiles) {
    __shared__ float lds[8][16 * 18];          // per-wave 16 rows x 18 (col0 = t0-1), pad kills conflicts
    int lane = threadIdx.x & 31;
    int wave = threadIdx.x >> 5;
    int tile = blockIdx.x * 8 + wave;
    if (tile >= tiles) tile = tiles - 1;       // clamp (idempotent rewrite) so barrier is uniform
    int nT = T >> 4;
    int tb = tile / nT;                        // batch tile
    int tt = tile - tb * nT;                   // time tile
    int t0 = tt << 4;

    float b0 = bc[0], b1 = bc[1];
    float a1 = ac[0], a2 = ac[1];

    float* tl = &lds[wave][0];

    // Coalesced tile load: each instruction covers two contiguous 64B row segments.
    #pragma unroll
    for (int c = 0; c < 8; ++c) {
        int flat = c * 32 + lane;              // 0..255
        int rr = flat >> 4, cc = flat & 15;
        tl[rr * 18 + 1 + cc] = u[(size_t)(tb * 16 + rr) * T + t0 + cc];
    }
    // Previous-sample column (t0-1); at t0==0 take u[-1] from u_init. Branchless ptr select.
    if (lane < 16) {
        int rowg = tb * 16 + lane;
        const float* p = (t0 == 0) ? (u_init + (size_t)rowg * 2)
                                   : (u + (size_t)rowg * T + (t0 - 1));
        tl[lane * 18] = *p;
    }
    __syncthreads();

    int m    = lane & 15;                      // M for A-frag, N for B/D-frags
    int half = lane >> 4;

    float hl = h_capture_lane(a1, a2, lane);   // lane n holds h[n]

    // A (16x4 f32 per slice): lanes 0-15 hold K=0,1; lanes 16-31 hold K=2,3.
    // B mirrored: B[K,N] = h[N - kglobal] (lower-triangular Toeplitz), via ds_bpermute.
    v2f Af[4]; v2f Bf[4];
#pragma unroll
    for (int j = 0; j < 4; ++j) {
#pragma unroll
        for (int r = 0; r < 2; ++r) {
            int k = 4 * j + 2 * half + r;
            float uc = tl[m * 18 + 1 + k];     // u[t0+k]
            float up = tl[m * 18 + k];         // u[t0+k-1]
            Af[j][r] = b0 * uc + b1 * up;      // FIR part x[t]
            Bf[j][r] = h_gather(hl, m - k);
        }
    }

    v8f acc = {0.f, 0.f, 0.f, 0.f, 0.f, 0.f, 0.f, 0.f};
#pragma unroll
    for (int j = 0; j < 4; ++j)
        acc = __builtin_amdgcn_wmma_f32_16x16x4_f32(
            /*neg_a=*/false, Af[j], /*neg_b=*/false, Bf[j],
            /*c_mod=*/(short)0, acc, /*reuse_a=*/false, /*reuse_b=*/false);

    // D layout: lane holds N = m, rows M = 8*half + r in VGPR r.
    float* orow = y0 + (size_t)(tb * 16 + 8 * half) * T + t0 + m;
#pragma unroll
    for (int r = 0; r < 8; ++r)
        orow[(size_t)r * T] = acc[r];
}

// Kernel 2: per-row sequential scan over tiles (chain shortened 16x by kernel 1).
//   s_start(c+1) = M^16 * s_start(c) + (y0[c,15], y0[c,14]),  M^16 from h[14..16].
__global__ void iir_state_scan(const float* __restrict__ y0,
                               const float* __restrict__ y_init,
                               const float* __restrict__ ac,
                               float2* __restrict__ states,   // (B, nT)
                               int B, int T) {
    int b = blockIdx.x * blockDim.x + threadIdx.x;
    if (b >= B) return;
    float a1 = ac[0], a2 = ac[1];
    float h[17];
    h[0] = 1.0f; h[1] = -a1;
#pragma unroll
    for (int i = 2; i <= 16; ++i) h[i] = -a1 * h[i - 1] - a2 * h[i - 2];
    float m00 = h[16], m01 = -a2 * h[15], m10 = h[15], m11 = -a2 * h[14];   // M^16
    float s1 = y_init[(size_t)b * 2 + 0];   // y[-1]
    float s2 = y_init[(size_t)b * 2 + 1];   // y[-2]
    int nT = T >> 4;
    const float* rowp = y0 + (size_t)b * T;
    float2* st = states + (size_t)b * nT;
    for (int c = 0; c < nT; ++c) {
        st[c] = make_float2(s1, s2);
        float d0 = rowp[c * 16 + 15];
        float d1 = rowp[c * 16 + 14];
        float n1 = m00 * s1 + m01 * s2 + d0;
        float n2 = m10 * s1 + m11 * s2 + d1;
        s1 = n1; s2 = n2;
    }
}

// Kernel 3: add zero-input response of each tile, in place on d_out.
//   y[b, 16c+j] += h[j+1]*s1 - a2*h[j]*s2   (h[.] gathered via ds_bpermute, branchless)
__global__ void iir_apply_state(float* __restrict__ y,
                                const float2* __restrict__ states,
                                const float* __restrict__ ac,
                                int T, size_t total) {
    size_t idx = (size_t)blockIdx.x * blockDim.x + threadIdx.x;
    if (idx >= total) return;                  // total % 256 == 0 -> wave-uniform in practice
    size_t b = idx / (size_t)T;
    int t = (int)(idx - b * (size_t)T);
    int j = t & 15;
    int c = t >> 4;
    float a1 = ac[0], a2 = ac[1];
    int lane = threadIdx.x & 31;
    float hl = h_capture_lane(a1, a2, lane);   // lane n holds h[n], n<=16
    float g1 = h_gather(hl, j + 1);            // h[j+1]
    float g2 = h_gather(hl, j);                // h[j]
    float2 s = states[b * (size_t)(T >> 4) + c];
    y[idx] += g1 * s.x + (-a2 * g2) * s.y;
}

extern "C" void kernel_launch(void* const* d_in, const int* in_sizes, int n_in,
                              void* d_out, int out_size, void* d_ws, size_t ws_size,
                              hipStream_t stream) {
    const float* bc = (const float*)d_in[0];   // b_coeff (2)
    const float* ac = (const float*)d_in[1];   // a_coeff (2)
    const float* u  = (const float*)d_in[2];   // u_in (B*T)
    const float* yi = (const float*)d_in[3];   // y_init (B*2)
    const float* ui = (const float*)d_in[4];   // u_init (B*2)
    float* y = (float*)d_out;

    int B = in_sizes[3] / 2;
    int T = in_sizes[2] / B;
    int nT = T >> 4;
    int tiles = (B >> 4) * nT;
    float2* states = (float2*)d_ws;            // B * nT * 8 bytes = 8 MB

    int blocks1 = (tiles + 7) / 8;             // 8 wave32 per block, one tile per wave
    iir_tile_fir_wmma<<<blocks1, 256, 0, stream>>>(u, ui, bc, ac, y, T, tiles);

    iir_state_scan<<<(B + 255) / 256, 256, 0, stream>>>(y, yi, ac, states, B, T);

    size_t total = (size_t)B * T;
    iir_apply_state<<<(int)((total + 255) / 256), 256, 0, stream>>>(y, states, ac, T, total);
}